// AdditiveAttention_68719477307
// MI455X (gfx1250) — compile-verified
//
#include <hip/hip_runtime.h>

// ---------------------------------------------------------------------------
// Additive attention, MI455X (gfx1250, wave32)
//   B=16, Q=K=H=DV=256
//   proj (WMMA f32 16x16x4)  ->  tanh-score (VALU/TRANS)  ->  softmax
//   ->  attn @ V (WMMA f32 16x16x4)
// Tile staging uses CDNA5 async global->LDS (ASYNCcnt) instead of
// load+ds_store round-trips.
// ---------------------------------------------------------------------------

typedef __attribute__((ext_vector_type(2))) float v2f;
typedef __attribute__((ext_vector_type(8))) float v8f;

#define HH      256
#define QTILE   16
#define LDSPAD  260   // row stride in floats: mult of 4 (16B align), %64==4 (bank spread)

// ---- tanh: prefer hardware v_tanh_f32 (TRANS) ------------------------------
#if __has_builtin(__builtin_amdgcn_tanhf)
__device__ __forceinline__ float fast_tanh(float x) {
    return __builtin_amdgcn_tanhf(x);                    // v_tanh_f32
}
#else
#warning "gfx1250: __builtin_amdgcn_tanhf not available -> exp2 fallback path"
__device__ __forceinline__ float fast_tanh(float x) {
    float e = __builtin_amdgcn_exp2f(2.8853900817779268f * x);  // e^(2x)
    return 1.0f - 2.0f / (e + 1.0f);
}
#endif

__device__ __forceinline__ float fast_exp(float x) {
    return __builtin_amdgcn_exp2f(1.4426950408889634f * x);     // v_exp_f32
}

// ---- CDNA5 async global->LDS copy (16B per lane, ASYNCcnt tracked) ---------
__device__ __forceinline__ void async_ld_b128(float* lds_ptr, const float* gptr) {
    unsigned loff = (unsigned)(uintptr_t)lds_ptr;   // LDS aperture: offset = addr[31:0]
    asm volatile("global_load_async_to_lds_b128 %0, %1, off"
                 :: "v"(loff), "v"(gptr) : "memory");
}
__device__ __forceinline__ void wait_async0() {
    asm volatile("s_wait_asynccnt 0x0" ::: "memory");
}

// Stage a 16x256 f32 tile (row-major, LDSPAD-padded rows) from global.
// 1024 16B chunks, 4 per thread (blockDim = 256). Coalesced.
__device__ __forceinline__ void stage_tile_async(float* dst_sm, const float* src) {
    const int tid = threadIdx.x;
    #pragma unroll
    for (int j = 0; j < 4; ++j) {
        int idx = j * 256 + tid;          // chunk id
        int r = idx >> 6;                 // 64 chunks per row
        int c = idx & 63;
        async_ld_b128(&dst_sm[r * LDSPAD + 4 * c], &src[r * HH + 4 * c]);
    }
}

// ---------------------------------------------------------------------------
// Y[M,256] = X[M,256] @ W^T   (W is [256,256] row-major, nn.Linear convention)
// One workgroup (8 waves) per 16-row M tile; each wave does 2 of 16 N tiles.
// ---------------------------------------------------------------------------
__global__ __launch_bounds__(256) void proj_kernel(const float* __restrict__ X,
                                                   const float* __restrict__ W,
                                                   float* __restrict__ Y) {
    __shared__ float a_sm[QTILE * LDSPAD];

    const int m0  = blockIdx.x * QTILE;
    const int tid = threadIdx.x;

    stage_tile_async(a_sm, &X[m0 * HH]);
    wait_async0();
    __syncthreads();

    const int lane = tid & 31;
    const int wave = tid >> 5;
    const int mn   = lane & 15;               // M index for A frag, N index for B frag
    const int koff = (lane < 16) ? 0 : 2;     // lanes 16-31 hold K+2 (ISA 16x4 f32 layout)

    for (int nt = wave; nt < 16; nt += 8) {
        const int n = nt * 16 + mn;
        v8f acc = {};
        for (int k0 = 0; k0 < HH; k0 += 4) {
            v2f a, b;
            const float* ar = &a_sm[mn * LDSPAD + k0 + koff];
            a.x = ar[0];  a.y = ar[1];                  // A[m][k], A[m][k+1]
            const float* wp = &W[n * HH + k0 + koff];   // B[k][n] = W[n][k]
            b.x = wp[0];  b.y = wp[1];
            acc = __builtin_amdgcn_wmma_f32_16x16x4_f32(false, a, false, b,
                                                        (short)0, acc, false, false);
        }
        // C/D layout: VGPR i, lanes 0-15 -> M=i ; lanes 16-31 -> M=i+8 ; N=lane&15
        const int mrow0 = (lane < 16) ? 0 : 8;
        #pragma unroll
        for (int i = 0; i < 8; ++i)
            Y[(m0 + mrow0 + i) * HH + n] = acc[i];
    }
}

// ---------------------------------------------------------------------------
// Fused: tanh-scores + softmax + attn@V.
// grid = (Q/16, B); block = 256 threads (8 wave32).
// ---------------------------------------------------------------------------
__global__ __launch_bounds__(256) void attn_kernel(const float* __restrict__ qp,
                                                   const float* __restrict__ kp,
                                                   const float* __restrict__ values,
                                                   const float* __restrict__ w_v,
                                                   float* __restrict__ out) {
    __shared__ float q_sm[QTILE * LDSPAD];
    __shared__ float k_sm[QTILE * LDSPAD];
    __shared__ float s_sm[QTILE * LDSPAD];
    __shared__ float wv_sm[HH];

    const int tid = threadIdx.x;
    const int qt  = blockIdx.x;
    const int bb  = blockIdx.y;
    const int q0  = qt * QTILE;

    wv_sm[tid] = w_v[tid];
    stage_tile_async(q_sm, &qp[((bb * 256) + q0) * HH]);
    wait_async0();
    __syncthreads();

    const int q  = tid >> 4;   // 0..15 : q row within tile
    const int kk = tid & 15;   // 0..15 : k within k-block

    // ---- scores: s[q][k] = sum_h w[h] * tanh(qp+kp) -------------------------
    for (int kb = 0; kb < 16; ++kb) {
        stage_tile_async(k_sm, &kp[((bb * 256) + kb * 16) * HH]);
        wait_async0();
        __syncthreads();

        float s = 0.0f;
        const float* qrow = &q_sm[q  * LDSPAD];
        const float* krow = &k_sm[kk * LDSPAD];
        for (int h = 0; h < HH; h += 4) {
            float4 qv = *reinterpret_cast<const float4*>(qrow + h);
            float4 kv = *reinterpret_cast<const float4*>(krow + h);
            s += wv_sm[h + 0] * fast_tanh(qv.x + kv.x);
            s += wv_sm[h + 1] * fast_tanh(qv.y + kv.y);
            s += wv_sm[h + 2] * fast_tanh(qv.z + kv.z);
            s += wv_sm[h + 3] * fast_tanh(qv.w + kv.w);
        }
        s_sm[q * LDSPAD + kb * 16 + kk] = s;
        __syncthreads();
    }

    // ---- softmax over K=256 per q row (16 lanes per row, wave32 butterfly) --
    float r[16];
    float mx = -1e30f;
    #pragma unroll
    for (int j = 0; j < 16; ++j) {
        r[j] = s_sm[q * LDSPAD + kk + 16 * j];
        mx = fmaxf(mx, r[j]);
    }
    #pragma unroll
    for (int msk = 1; msk < 16; msk <<= 1)
        mx = fmaxf(mx, __shfl_xor(mx, msk, 32));
    float sum = 0.0f;
    #pragma unroll
    for (int j = 0; j < 16; ++j) { r[j] = fast_exp(r[j] - mx); sum += r[j]; }
    #pragma unroll
    for (int msk = 1; msk < 16; msk <<= 1)
        sum += __shfl_xor(sum, msk, 32);
    const float inv = 1.0f / sum;
    #pragma unroll
    for (int j = 0; j < 16; ++j)
        s_sm[q * LDSPAD + kk + 16 * j] = r[j] * inv;
    __syncthreads();

    // ---- out[16,256] = attn[16,256] @ V[b][256,256] via WMMA f32 16x16x4 ----
    const int lane = tid & 31;
    const int wave = tid >> 5;
    const int mn   = lane & 15;
    const int koff = (lane < 16) ? 0 : 2;

    for (int nt = wave; nt < 16; nt += 8) {
        const int n = nt * 16 + mn;
        v8f acc = {};
        for (int k0 = 0; k0 < 256; k0 += 4) {
            v2f a, bm;
            const float* ar = &s_sm[mn * LDSPAD + k0 + koff];
            a.x = ar[0];  a.y = ar[1];                         // attn[m][k..k+1]
            const float* vp = &values[((bb * 256) + k0 + koff) * 256 + n];
            bm.x = vp[0];                                      // V[k][n]
            bm.y = vp[256];                                    // V[k+1][n]
            acc = __builtin_amdgcn_wmma_f32_16x16x4_f32(false, a, false, bm,
                                                        (short)0, acc, false, false);
        }
        const int mrow0 = (lane < 16) ? 0 : 8;
        #pragma unroll
        for (int i = 0; i < 8; ++i)
            out[((bb * 256) + q0 + mrow0 + i) * 256 + n] = acc[i];
    }
}

// ---------------------------------------------------------------------------
extern "C" void kernel_launch(void* const* d_in, const int* in_sizes, int n_in,
                              void* d_out, int out_size, void* d_ws, size_t ws_size,
                              hipStream_t stream) {
    const float* queries = (const float*)d_in[0];
    const float* keys    = (const float*)d_in[1];
    const float* values  = (const float*)d_in[2];
    const float* W_q     = (const float*)d_in[3];
    const float* W_k     = (const float*)d_in[4];
    const float* w_v     = (const float*)d_in[5];

    float* qproj = (float*)d_ws;                       // [16*256, 256] = 4 MB
    float* kproj = qproj + 16 * 256 * 256;             // [16*256, 256] = 4 MB

    proj_kernel<<<dim3(256), dim3(256), 0, stream>>>(queries, W_q, qproj);
    proj_kernel<<<dim3(256), dim3(256), 0, stream>>>(keys,    W_k, kproj);
    attn_kernel<<<dim3(16, 16), dim3(256), 0, stream>>>(qproj, kproj, values, w_v,
                                                        (float*)d_out);
}